// GCNLayer_34351148433457
// MI455X (gfx1250) — compile-verified
//
#include <hip/hip_runtime.h>
#include <hip/hip_bf16.h>

// GCN layer: out = relu( (segsum(order * (h@W)[src] -> dst) / max(deg,1)) + b )
// Sizes: N=100000 nodes, E=1600000 edges, IN=64, OUT=32.

typedef __attribute__((ext_vector_type(2))) float v2f;
typedef __attribute__((ext_vector_type(8))) float v8f;

#define IN_F 64
#define OUT_F 32

// ---------------------------------------------------------------------------
// Zero fill (d_out and deg must start at 0 each call; ws/d_out are poisoned).
// ---------------------------------------------------------------------------
__global__ void zero_f32(float* __restrict__ p, int n) {
    int i = blockIdx.x * blockDim.x + threadIdx.x;
    if (i < n) p[i] = 0.0f;
}

// ---------------------------------------------------------------------------
// hW = h @ W  via V_WMMA_F32_16X16X4_F32 (fp32-exact, matches reference).
// One wave -> one 16-row x 32-col output tile: two 16x16 accumulators,
// 16 k-steps of K=4 => 32 wmma instructions per wave.
//
// VGPR layouts (ISA 7.12.2):
//   A (16x4):  lane%16 = M;  vgpr v holds K = kb + 2*(lane/16) + v
//   B (4x16):  lane%16 = N;  vgpr v holds K = kb + 2*(lane/16) + v
//   C (16x16): lane%16 = N;  vgpr v holds M = v + 8*(lane/16)
// ---------------------------------------------------------------------------
__global__ void gemm_hW_wmma(const float* __restrict__ h,
                             const float* __restrict__ W,
                             float* __restrict__ hW,
                             int n_tiles) {
    int wave = (blockIdx.x * blockDim.x + threadIdx.x) >> 5;
    if (wave >= n_tiles) return;            // uniform per wave: EXEC stays all-1
    int lane = threadIdx.x & 31;
    int half = lane >> 4;                   // 0: lanes 0-15, 1: lanes 16-31
    int l16  = lane & 15;

    const float* __restrict__ hrow = h + (size_t)(wave * 16 + l16) * IN_F;

    v8f c0 = {};                            // N = 0..15
    v8f c1 = {};                            // N = 16..31

#pragma unroll
    for (int kb = 0; kb < IN_F; kb += 4) {
        int k0 = kb + 2 * half;
        // A fragment: float2 load of h[row][k0], h[row][k0+1]  (one b64 load)
        v2f a = *(const v2f*)(hrow + k0);
        // B fragments for the two N-tiles (W is 64x32, tiny -> L0/L2 resident)
        v2f b0, b1;
        b0.x = W[(size_t)k0       * OUT_F + l16];
        b0.y = W[(size_t)(k0 + 1) * OUT_F + l16];
        b1.x = W[(size_t)k0       * OUT_F + 16 + l16];
        b1.y = W[(size_t)(k0 + 1) * OUT_F + 16 + l16];

        c0 = __builtin_amdgcn_wmma_f32_16x16x4_f32(false, a, false, b0,
                                                   (short)0, c0, false, false);
        c1 = __builtin_amdgcn_wmma_f32_16x16x4_f32(false, a, false, b1,
                                                   (short)0, c1, false, false);
    }

    // Store D: vgpr v -> row (wave*16 + v + 8*half), col l16 / l16+16
    int mbase = wave * 16 + 8 * half;
#pragma unroll
    for (int v = 0; v < 8; ++v) {
        hW[(size_t)(mbase + v) * OUT_F + l16]      = c0[v];
        hW[(size_t)(mbase + v) * OUT_F + 16 + l16] = c1[v];
    }
}

// ---------------------------------------------------------------------------
// Edge scatter: one wave per edge, one lane per output feature.
//   - src/dst/order loads are wave-uniform (coalesce to one transaction)
//   - hW[src] row read: 32 lanes x 4B contiguous = 128B (L2-resident, 12.8MB)
//   - agg[dst] atomic add: 32 contiguous global_atomic_add_f32 (L2 atomics)
//   - lane 0 bumps deg[dst]
// ---------------------------------------------------------------------------
__global__ void edge_scatter(const int* __restrict__ src,
                             const int* __restrict__ dst,
                             const float* __restrict__ order,
                             const float* __restrict__ hW,
                             float* __restrict__ agg,
                             float* __restrict__ deg,
                             int n_edges) {
    unsigned tid = blockIdx.x * blockDim.x + threadIdx.x;
    int e = (int)(tid >> 5);
    if (e >= n_edges) return;               // uniform per wave
    int f = (int)(tid & 31);

    int   s = src[e];
    int   d = dst[e];
    float w = order[e];

    float m = w * hW[(size_t)s * OUT_F + f];
    atomicAdd(&agg[(size_t)d * OUT_F + f], m);
    if (f == 0) atomicAdd(&deg[d], 1.0f);
}

// ---------------------------------------------------------------------------
// Finalize in place: out = relu(agg * 1/max(deg,1) + b)
// ---------------------------------------------------------------------------
__global__ void finalize(float* __restrict__ out,
                         const float* __restrict__ deg,
                         const float* __restrict__ b,
                         int n_nodes) {
    int tid = blockIdx.x * blockDim.x + threadIdx.x;
    if (tid >= n_nodes * OUT_F) return;
    int n = tid >> 5;                       // OUT_F == 32
    int f = tid & 31;
    float norm = 1.0f / fmaxf(deg[n], 1.0f);
    float v = out[tid] * norm + b[f];
    out[tid] = fmaxf(v, 0.0f);
}

// ---------------------------------------------------------------------------
// Launch
// ---------------------------------------------------------------------------
extern "C" void kernel_launch(void* const* d_in, const int* in_sizes, int n_in,
                              void* d_out, int out_size, void* d_ws, size_t ws_size,
                              hipStream_t stream) {
    const float* h     = (const float*)d_in[0];   // [N, 64]
    const int*   src   = (const int*)  d_in[1];   // [E]
    const int*   dst   = (const int*)  d_in[2];   // [E]
    const float* order = (const float*)d_in[3];   // [E]
    const float* W     = (const float*)d_in[4];   // [64, 32]
    const float* b     = (const float*)d_in[5];   // [32]

    const int n_nodes = in_sizes[0] / IN_F;       // 100000
    const int n_edges = in_sizes[1];              // 1600000

    float* out = (float*)d_out;                   // [N, 32] fp32
    float* hW  = (float*)d_ws;                    // [N, 32] = 12.8 MB
    float* deg = hW + (size_t)n_nodes * OUT_F;    // [N]     =  0.4 MB

    const int B = 256;

    // 1) zero the accumulation target (d_out) and the degree counters
    {
        int n = n_nodes * OUT_F;
        zero_f32<<<(n + B - 1) / B, B, 0, stream>>>(out, n);
        zero_f32<<<(n_nodes + B - 1) / B, B, 0, stream>>>(deg, n_nodes);
    }

    // 2) hW = h @ W  (WMMA fp32). 16 rows per wave -> n_tiles waves.
    {
        int n_tiles = (n_nodes + 15) / 16;        // 6250
        int waves_per_block = B / 32;             // 8
        int blocks = (n_tiles + waves_per_block - 1) / waves_per_block;
        gemm_hW_wmma<<<blocks, B, 0, stream>>>(h, W, hW, n_tiles);
    }

    // 3) per-edge gather/scale/scatter-add (+degree count)
    {
        long long threads = (long long)n_edges * 32;
        int blocks = (int)((threads + B - 1) / B);
        edge_scatter<<<blocks, B, 0, stream>>>(src, dst, order, hW, out, deg, n_edges);
    }

    // 4) normalize + bias + relu, in place on d_out
    {
        int n = n_nodes * OUT_F;
        finalize<<<(n + B - 1) / B, B, 0, stream>>>(out, deg, b, n_nodes);
    }
}